// JosephProjector3D_34316788695355
// MI455X (gfx1250) — compile-verified
//
#include <hip/hip_runtime.h>
#include <math.h>

// Geometry (compile-time, matches reference)
#define DD 256
#define HH 256
#define WW 256
#define AA 48
#define UU 256
#define VV 256
#define NSTEPS 96

typedef float  f4 __attribute__((ext_vector_type(4)));
typedef int    i4 __attribute__((ext_vector_type(4)));

__device__ __forceinline__ f4 ldvec(const float* p) {
    return *(const f4*)p;
}

__device__ __forceinline__ void accum(f4& acc, const f4 w,
                                      const f4 c00, const f4 c01,
                                      const f4 c10, const f4 c11) {
    acc.x = fmaf(w.x, c00.x, fmaf(w.y, c01.x, fmaf(w.z, c10.x, fmaf(w.w, c11.x, acc.x))));
    acc.y = fmaf(w.x, c00.y, fmaf(w.y, c01.y, fmaf(w.z, c10.y, fmaf(w.w, c11.y, acc.y))));
    acc.z = fmaf(w.x, c00.z, fmaf(w.y, c01.z, fmaf(w.z, c10.z, fmaf(w.w, c11.z, acc.z))));
    acc.w = fmaf(w.x, c00.w, fmaf(w.y, c01.w, fmaf(w.z, c10.w, fmaf(w.w, c11.w, acc.w))));
}

// Inline sincos for small args (|a| < 4). Angles here are k*pi/48 in [0, pi):
// one Cody-Waite reduction by pi/2 (k in {0,1,2}) + minimax polys (~1 ulp on
// |r| <= pi/4). Lowered by the compiler to gfx1250 scalar-float SALU ops.
__device__ __forceinline__ void sincos_small(float a, float* sp, float* cp) {
    const float TWO_OPI  = 0.636619772367581343f;   // 2/pi
    const float PIO2_HI  = 1.57079637050628662109f; // float(pi/2)
    const float PIO2_LO  = -4.37113900018624283e-8f;
    const float kf = rintf(a * TWO_OPI);
    const int   k  = (int)kf;
    float r = fmaf(-kf, PIO2_HI, a);
    r       = fmaf(-kf, PIO2_LO, r);
    const float r2 = r * r;
    const float sr = fmaf(r * r2,
        fmaf(r2, fmaf(r2, -1.9840874e-4f, 8.3333293e-3f), -1.66666672e-1f), r);
    const float cr = fmaf(r2,
        fmaf(r2, fmaf(r2, -1.3888397e-3f, 4.1666418e-2f), -0.5f), 1.0f);
    float sv = (k & 1) ? cr : sr;
    float cv = (k & 1) ? -sr : cr;
    if (k & 2) { sv = -sv; cv = -cv; }
    *sp = sv;
    *cp = cv;
}

// One block per (u, a) ray-column; 64 threads, each owning 4 consecutive v
// (v contiguous in the input's [X,Y,Z] layout -> coalesced b128 gathers with
// wave-uniform bases). z-interp is exact plane select (fz == 0).
__global__ __launch_bounds__(64)
void joseph_fwd_kernel(const float* __restrict__ vol,
                       const float* __restrict__ angles,
                       float* __restrict__ out) {
    const int u   = blockIdx.x;
    const int a   = blockIdx.y;
    const int tid = threadIdx.x;

    __shared__ i4 s_b[NSTEPS];   // 4 corner base offsets (x*H+y)*D
    __shared__ f4 s_w[NSTEPS];   // 4 masked bilinear weights

    const float T  = 180.31222920256963f;        // 0.5*sqrt(2)*255
    const float DT = 2.0f * T / (float)NSTEPS;   // Joseph step length

    float s, c;
    sincos_small(angles[a], &s, &c);
    const float uc = (float)u - 127.5f;
    const float x0 = -uc * s;
    const float y0 =  uc * c;

    // --- Precompute per-step uniform params into LDS (96 steps / 64 thr) ---
    for (int i = tid; i < NSTEPS; i += 64) {
        const float t  = -T + ((float)i + 0.5f) * DT;
        const float xf = x0 + t * c + 127.5f;
        const float yf = y0 + t * s + 127.5f;
        const float xfl = floorf(xf);
        const float yfl = floorf(yf);
        const float fx = xf - xfl;
        const float fy = yf - yfl;
        const int ix0 = (int)xfl, iy0 = (int)yfl;
        const int ix1 = ix0 + 1,  iy1 = iy0 + 1;
        const float mx0 = (ix0 >= 0 && ix0 < WW) ? 1.0f : 0.0f;
        const float mx1 = (ix1 >= 0 && ix1 < WW) ? 1.0f : 0.0f;
        const float my0 = (iy0 >= 0 && iy0 < HH) ? 1.0f : 0.0f;
        const float my1 = (iy1 >= 0 && iy1 < HH) ? 1.0f : 0.0f;
        const int cx0 = min(max(ix0, 0), WW - 1);
        const int cx1 = min(max(ix1, 0), WW - 1);
        const int cy0 = min(max(iy0, 0), HH - 1);
        const int cy1 = min(max(iy1, 0), HH - 1);
        const float ax0 = (1.0f - fx) * mx0;
        const float ax1 = fx * mx1;
        const float ay0 = (1.0f - fy) * my0;
        const float ay1 = fy * my1;
        f4 w; w.x = ay0 * ax0; w.y = ay0 * ax1; w.z = ay1 * ax0; w.w = ay1 * ax1;
        i4 b;
        b.x = (cx0 * HH + cy0) * DD;   // (yi0, xi0)
        b.y = (cx1 * HH + cy0) * DD;   // (yi0, xi1)
        b.z = (cx0 * HH + cy1) * DD;   // (yi1, xi0)
        b.w = (cx1 * HH + cy1) * DD;   // (yi1, xi1)
        s_w[i] = w;
        s_b[i] = b;
    }
    __syncthreads();

    const int v4 = tid * 4;            // this thread's first v (z) index
    f4 acc; acc.x = 0.f; acc.y = 0.f; acc.z = 0.f; acc.w = 0.f;

    // Warm the pipeline: stage A <- step 0; prefetch step 1 lines.
    __builtin_prefetch(vol + s_b[1].x + v4, 0, 1);   // global_prefetch_b8
    __builtin_prefetch(vol + s_b[1].w + v4, 0, 1);
    f4 wA = s_w[0];
    {
        const i4 bA = s_b[0];
        // loads issued immediately below
        f4 dummy; (void)dummy;
        // fallthrough
        // (loads declared outside the block)
        // -- see below
        (void)bA;
    }
    i4 bA_ = s_b[0];
    f4 a00 = ldvec(vol + bA_.x + v4);
    f4 a01 = ldvec(vol + bA_.y + v4);
    f4 a10 = ldvec(vol + bA_.z + v4);
    f4 a11 = ldvec(vol + bA_.w + v4);

    // --- Ping-pong pipelined loop, 2 steps per iteration (no reg rotation) ---
    // Iter i: fetch B <- step i+1; accum A (step i); fetch A <- step i+2;
    //         accum B (step i+1).   NSTEPS is even; last pair peeled.
    for (int i = 0; i < NSTEPS - 2; i += 2) {
        const i4 bB = s_b[i + 1];
        const f4 wB = s_w[i + 1];
        f4 b00 = ldvec(vol + bB.x + v4);
        f4 b01 = ldvec(vol + bB.y + v4);
        f4 b10 = ldvec(vol + bB.z + v4);
        f4 b11 = ldvec(vol + bB.w + v4);

        accum(acc, wA, a00, a01, a10, a11);          // step i

        const i4 bA = s_b[i + 2];
        wA = s_w[i + 2];
        a00 = ldvec(vol + bA.x + v4);
        a01 = ldvec(vol + bA.y + v4);
        a10 = ldvec(vol + bA.z + v4);
        a11 = ldvec(vol + bA.w + v4);

        accum(acc, wB, b00, b01, b10, b11);          // step i+1
    }
    {   // epilogue: steps NSTEPS-2 (in A) and NSTEPS-1
        const i4 bB = s_b[NSTEPS - 1];
        const f4 wB = s_w[NSTEPS - 1];
        const f4 b00 = ldvec(vol + bB.x + v4);
        const f4 b01 = ldvec(vol + bB.y + v4);
        const f4 b10 = ldvec(vol + bB.z + v4);
        const f4 b11 = ldvec(vol + bB.w + v4);
        accum(acc, wA, a00, a01, a10, a11);
        accum(acc, wB, b00, b01, b10, b11);
    }

    acc.x *= DT; acc.y *= DT; acc.z *= DT; acc.w *= DT;

    // out layout [B,C,U,A,V] -> flat (u*A + a)*V + v ; write-once, stream it
    // non-temporally so the 64MB volume stays L2-resident.
    float* op = out + ((size_t)u * AA + a) * VV + v4;
    __builtin_nontemporal_store(acc, (f4*)op);
}

extern "C" void kernel_launch(void* const* d_in, const int* in_sizes, int n_in,
                              void* d_out, int out_size, void* d_ws, size_t ws_size,
                              hipStream_t stream) {
    const float* vol    = (const float*)d_in[0];   // [1,1,256,256,256] (X,Y,Z)
    const float* angles = (const float*)d_in[1];   // [48]
    float* out = (float*)d_out;                    // [1,1,256,48,256]  (U,A,V)

    dim3 grid(UU, AA, 1);
    dim3 block(64, 1, 1);
    joseph_fwd_kernel<<<grid, block, 0, stream>>>(vol, angles, out);
}